// BaseRouter_73031623901311
// MI455X (gfx1250) — compile-verified
//
#include <hip/hip_runtime.h>
#include <hip/hip_bf16.h>
#include <stdint.h>

// CDNA5 / gfx1250 MoE-style router: exact top-k (JAX top_k semantics) + row gather.
// Gather is HBM-bound (~128 MB moved -> ~5.5us at 23.3 TB/s); it uses the gfx1250
// async data mover (global_load_async_to_lds_b128 / global_store_async_from_lds_b128,
// ASYNCcnt) to stream rows through LDS without VGPR staging.

#define ROUTER_T 4096
#define ROUTER_K 2048   // T * CAPACITY(0.5)

typedef __attribute__((address_space(3))) float lds_float;

static __device__ __forceinline__ void wait_asynccnt0() {
#if __has_builtin(__builtin_amdgcn_s_wait_asynccnt)
    __builtin_amdgcn_s_wait_asynccnt(0);
#else
    asm volatile("s_wait_asynccnt 0x0" ::: "memory");
#endif
}

// -------- Phase 1: exact rank-select top-k (descending, stable ties -> lower index) -----
// One block (1024 threads) per batch row. Scores staged in LDS; each thread keeps 4
// candidates and accumulates their ranks in a single pass over the row.
__global__ __launch_bounds__(1024) void topk_select_kernel(
    const float* __restrict__ scores,
    float* __restrict__ out_batch,   // [B*K] batch index as float
    float* __restrict__ out_idx,     // [B*K] token index as float
    float* __restrict__ out_vals)    // [B*K] top-k values, sorted descending
{
    __shared__ float s[ROUTER_T];
    const int b = blockIdx.x;
    const float* row = scores + (size_t)b * ROUTER_T;

    for (int i = threadIdx.x; i < ROUTER_T; i += 1024) s[i] = row[i];
    __syncthreads();

    const int t = threadIdx.x;
    float v[4];
    int   idx[4];
    int   rank[4];
#pragma unroll
    for (int m = 0; m < 4; ++m) {
        idx[m]  = t + m * 1024;
        v[m]    = s[idx[m]];
        rank[m] = 0;
    }

#pragma unroll 4
    for (int j = 0; j < ROUTER_T; ++j) {
        const float u = s[j];
#pragma unroll
        for (int m = 0; m < 4; ++m) {
            // strictly-greater wins; equal value wins iff lower index (JAX tie-break)
            rank[m] += (u > v[m]) || ((u == v[m]) && (j < idx[m]));
        }
    }

#pragma unroll
    for (int m = 0; m < 4; ++m) {
        if (rank[m] < ROUTER_K) {
            const int o = b * ROUTER_K + rank[m];
            out_vals[o]  = v[m];
            out_idx[o]   = (float)idx[m];  // exact: idx < 4096
            out_batch[o] = (float)b;
        }
    }
}

// -------- Phase 2: gather selected rows via the gfx1250 async data mover --------------
// One block per selected row. Each thread moves two 16B chunks:
// global --(global_load_async_to_lds_b128)--> LDS --(global_store_async_from_lds_b128)--> global.
// Per-wave ASYNCcnt orders load-completion before the stores; no cross-thread deps.
__global__ __launch_bounds__(256) void gather_rows_kernel(
    const float* __restrict__ hidden,   // [B, T, D]
    const float* __restrict__ idx_f,    // [B*K] token indices (as float, exact)
    float* __restrict__ out,            // [B*K, D]
    int D)
{
    __shared__ float buf[2048];         // D floats = 8 KB per block

    const int j   = blockIdx.x;         // selected-row id, 0 .. B*K-1
    const int b   = j >> 11;            // j / ROUTER_K
    const int tok = (int)idx_f[j];

    const float* src = hidden + ((size_t)b * ROUTER_T + (size_t)tok) * (size_t)D;
    float*       dst = out + (size_t)j * (size_t)D;

    const int nchunks = D >> 2;         // 16B chunks (512 for D=2048)
    lds_float* lbase = (lds_float*)buf;

    // async load: global -> LDS (one b128 per chunk, per lane)
    for (int c = threadIdx.x; c < nchunks; c += 256) {
        lds_float*   lp = lbase + (c << 2);
        const float* gp = src + (c << 2);
        asm volatile("global_load_async_to_lds_b128 %0, %1, off"
                     :: "v"(lp), "v"(gp) : "memory");
    }
    wait_asynccnt0();   // all of this wave's loads have landed in LDS

    // async store: LDS -> global
    for (int c = threadIdx.x; c < nchunks; c += 256) {
        lds_float* lp = lbase + (c << 2);
        float*     gp = dst + (c << 2);
        asm volatile("global_store_async_from_lds_b128 %0, %1, off"
                     :: "v"(gp), "v"(lp) : "memory");
    }
    wait_asynccnt0();   // drain before wave retires (S_ENDPGM would also wait-idle)
}

extern "C" void kernel_launch(void* const* d_in, const int* in_sizes, int n_in,
                              void* d_out, int out_size, void* d_ws, size_t ws_size,
                              hipStream_t stream) {
    (void)n_in; (void)d_ws; (void)ws_size; (void)out_size;

    const float* scores = (const float*)d_in[0];  // [B, T]
    const float* hidden = (const float*)d_in[1];  // [B, T, D]
    float* out = (float*)d_out;

    const int BT = in_sizes[0];          // B*T
    const int D  = in_sizes[1] / BT;     // 2048
    const int B  = BT / ROUTER_T;        // 4

    // Output layout (flat, reference return order):
    //   [0, B*K*D)                      selected_hidden
    //   [B*K*D, +B*K)                   batch_idx (as float)
    //   [.. , +B*K)                     topk_idx  (as float)
    //   [.. , +B*K)                     topk_vals
    const size_t bk = (size_t)B * ROUTER_K;
    float* out_hidden = out;
    float* out_batch  = out + bk * (size_t)D;
    float* out_idx    = out_batch + bk;
    float* out_vals   = out_idx + bk;

    topk_select_kernel<<<B, 1024, 0, stream>>>(scores, out_batch, out_idx, out_vals);
    gather_rows_kernel<<<(int)bk, 256, 0, stream>>>(hidden, out_idx, out_hidden, D);
}